// MultiHeadRouter_26345329394138
// MI455X (gfx1250) — compile-verified
//
#include <hip/hip_runtime.h>
#include <hip/hip_bf16.h>

// ---------------------------------------------------------------------------
// MultiHeadRouter for MI455X (gfx1250, wave32)
//   B=2 L=4096 H=16 S=64 D=128, T=8192
//   Memory-bound: 67MB of x @ 23.3TB/s ~ 2.9us floor. Single fused pass.
//   WMMA: V_WMMA_F32_16X16X4_F32, TDM: tensor_load_to_lds for weights.
// ---------------------------------------------------------------------------

#define NB   2
#define NL   4096
#define NH   16
#define NS   64
#define ND   128
#define NT   (NB * NL)      // 8192 rows
#define ROWS 128            // rows per block (8 waves x 16 rows)
#define XSTR 132            // padded LDS row stride (floats) -> bank-conflict free
#define WSTR 132

typedef __attribute__((ext_vector_type(2))) float    v2f;
typedef __attribute__((ext_vector_type(8))) float    v8f;
typedef __attribute__((ext_vector_type(4))) unsigned v4u;
typedef __attribute__((ext_vector_type(8))) int      v8i;
typedef __attribute__((ext_vector_type(4))) int      v4i;

__global__ void __launch_bounds__(256)
router_zero_ws(unsigned* __restrict__ g_cnt, float* __restrict__ g_ssum) {
    int i = blockIdx.x * 256 + threadIdx.x;
    if (i < NH * NS) { g_cnt[i] = 0u; g_ssum[i] = 0.0f; }
}

__global__ void __launch_bounds__(256)
router_main(const float* __restrict__ x, const float* __restrict__ w,
            const float* __restrict__ bias,
            float* __restrict__ out_sg, float* __restrict__ out_idx,
            unsigned* __restrict__ g_cnt, float* __restrict__ g_ssum) {
    __shared__ float    ws_lds[NS * WSTR];   // weight[h]: 64 x 128, padded stride
    __shared__ float    xs[ROWS * XSTR];     // x rows:   128 x 128, padded stride
    __shared__ unsigned hist[NS];
    __shared__ float    sse[NS];

    const int tid = threadIdx.x;
    const int h   = blockIdx.y;
    const int t0  = blockIdx.x * ROWS;

    if (tid < NS) { hist[tid] = 0u; sse[tid] = 0.0f; }

    // ---- TDM: DMA weight[h] (64x128 f32 = 32KB) into LDS, padded 4dw/128dw ----
    if (tid < 32) {  // wave 0 only (uniform branch; TDM ignores EXEC)
        unsigned lds_addr = (unsigned)(size_t)(void*)ws_lds;  // low 32b = LDS offset
        unsigned long long ga =
            (unsigned long long)(size_t)(w + (size_t)h * NS * ND);
        v4u g0;
        g0.x = 1u;                                     // count=1, user descriptor
        g0.y = lds_addr;                               // lds_addr
        g0.z = (unsigned)(ga & 0xFFFFFFFFull);         // global_addr[31:0]
        g0.w = (unsigned)((ga >> 32) & 0x1FFFFFFull)   // global_addr[56:32]
             | (2u << 30);                             // type=2 (image)
        v8i g1;
        g1[0] = (int)((2u << 16)        // data_size = 4 bytes
                    | (1u << 20)        // pad_enable
                    | (6u << 22)        // pad_interval: 128 dwords
                    | (3u << 25));      // pad_amount:   4 dwords -> stride 132
        g1[1] = (int)(128u << 16);      // tensor_dim0 = 128 (bits 79:48, lo16)
        g1[2] = (int)(64u << 16);       // dim0 hi = 0 | tensor_dim1 = 64 (lo16)
        g1[3] = (int)(128u << 16);      // dim1 hi = 0 | tile_dim0 = 128
        g1[4] = (int)64u;               // tile_dim1 = 64, tile_dim2 = 0
        g1[5] = (int)128u;              // tensor_dim0_stride = 128 (lo32)
        g1[6] = 0;                      // stride0 hi | stride1 lo
        g1[7] = 0;
        v4i gz = {0, 0, 0, 0};          // 2D tensor: groups 2/3 unused
#if __clang_major__ >= 23
        v8i gz8 = {0, 0, 0, 0, 0, 0, 0, 0};
        __builtin_amdgcn_tensor_load_to_lds(g0, g1, gz, gz, gz8, 0);
#else
        __builtin_amdgcn_tensor_load_to_lds(g0, g1, gz, gz, 0);
#endif
        __builtin_amdgcn_s_wait_tensorcnt(0);
    }

    // ---- stage 128 rows of x: coalesced (32 lanes cover one 512B row) ----
#pragma unroll
    for (int j = 0; j < 16; ++j) {
        int idx = j * 256 + tid;          // 4096 float4 total
        int row = idx >> 5, c4 = idx & 31;
        float4 v = *(const float4*)(x + ((size_t)(t0 + row) * NH + h) * ND + c4 * 4);
        *(float4*)(xs + row * XSTR + c4 * 4) = v;
    }
    __syncthreads();

    // ---- per-wave: 16 rows x 64 states via WMMA f32 16x16x4, K=128 ----
    const int lane = tid & 31;
    const int wave = tid >> 5;
    const int half = lane >> 4;   // K-group / row-group selector
    const int lrow = lane & 15;   // A row / B,C column within tile
    const int wrow = wave * 16;

    v8f acc0 = {0,0,0,0,0,0,0,0};
    v8f acc1 = {0,0,0,0,0,0,0,0};
    v8f acc2 = {0,0,0,0,0,0,0,0};
    v8f acc3 = {0,0,0,0,0,0,0,0};

    const float* xrow  = xs     + (wrow + lrow) * XSTR + 2 * half;
    const float* wbase = ws_lds + lrow * WSTR          + 2 * half;
#pragma unroll
    for (int kk = 0; kk < 32; ++kk) {
        v2f a  = *(const v2f*)(xrow  + 4 * kk);
        v2f b0 = *(const v2f*)(wbase +  0 * WSTR + 4 * kk);
        v2f b1 = *(const v2f*)(wbase + 16 * WSTR + 4 * kk);
        v2f b2 = *(const v2f*)(wbase + 32 * WSTR + 4 * kk);
        v2f b3 = *(const v2f*)(wbase + 48 * WSTR + 4 * kk);
        acc0 = __builtin_amdgcn_wmma_f32_16x16x4_f32(false, a, false, b0, (short)0, acc0, false, false);
        acc1 = __builtin_amdgcn_wmma_f32_16x16x4_f32(false, a, false, b1, (short)0, acc1, false, false);
        acc2 = __builtin_amdgcn_wmma_f32_16x16x4_f32(false, a, false, b2, (short)0, acc2, false, false);
        acc3 = __builtin_amdgcn_wmma_f32_16x16x4_f32(false, a, false, b3, (short)0, acc3, false, false);
    }

    // bias per column (lane holds column lrow of each 16-wide N tile)
    float bv0 = bias[h * NS +  0 + lrow];
    float bv1 = bias[h * NS + 16 + lrow];
    float bv2 = bias[h * NS + 32 + lrow];
    float bv3 = bias[h * NS + 48 + lrow];

    // ---- softmax + argmax per row; row m = 8*half + v lives in one half-wave ----
    float persum0 = 0.f, persum1 = 0.f, persum2 = 0.f, persum3 = 0.f;
#pragma unroll
    for (int v = 0; v < 8; ++v) {
        float v0 = acc0[v] + bv0;
        float v1 = acc1[v] + bv1;
        float v2 = acc2[v] + bv2;
        float v3 = acc3[v] + bv3;
        // in-lane max/argmax (scan ascending s -> keeps lowest index on ties)
        float m = v0; int arg = lrow;
        if (v1 > m) { m = v1; arg = 16 + lrow; }
        if (v2 > m) { m = v2; arg = 32 + lrow; }
        if (v3 > m) { m = v3; arg = 48 + lrow; }
        // butterfly across the 16-lane half (masks 1,2,4,8 stay within half)
#pragma unroll
        for (int mk = 1; mk <= 8; mk <<= 1) {
            float om = __shfl_xor(m, mk, 32);
            int   oa = __shfl_xor(arg, mk, 32);
            if (om > m || (om == m && oa < arg)) { m = om; arg = oa; }
        }
        float e0 = __expf(v0 - m), e1 = __expf(v1 - m);
        float e2 = __expf(v2 - m), e3 = __expf(v3 - m);
        float ps = e0 + e1 + e2 + e3;
#pragma unroll
        for (int mk = 1; mk <= 8; mk <<= 1) ps += __shfl_xor(ps, mk, 32);
        float inv = 1.0f / ps;
        persum0 += e0 * inv; persum1 += e1 * inv;
        persum2 += e2 * inv; persum3 += e3 * inv;

        if (lrow == 0) {
            int trow = t0 + wrow + half * 8 + v;
            out_sg [(size_t)trow * NH + h] = 1.0f;        // GradTop1 forward == 1
            out_idx[(size_t)trow * NH + h] = (float)arg;  // argmax index
            atomicAdd(&hist[arg], 1u);
        }
    }
    atomicAdd(&sse[ 0 + lrow], persum0);
    atomicAdd(&sse[16 + lrow], persum1);
    atomicAdd(&sse[32 + lrow], persum2);
    atomicAdd(&sse[48 + lrow], persum3);
    __syncthreads();

    if (tid < NS) {
        atomicAdd(&g_cnt [h * NS + tid], hist[tid]);
        atomicAdd(&g_ssum[h * NS + tid], sse[tid]);
    }
}

__global__ void __launch_bounds__(256)
router_finalize(const unsigned* __restrict__ g_cnt, const float* __restrict__ g_ssum,
                float* __restrict__ out_loss) {
    __shared__ float red[256];
    float p = 0.0f;
    for (int i = threadIdx.x; i < NH * NS; i += 256)
        p += (float)g_cnt[i] * g_ssum[i];
    red[threadIdx.x] = p;
    __syncthreads();
    for (int s = 128; s > 0; s >>= 1) {
        if (threadIdx.x < s) red[threadIdx.x] += red[threadIdx.x + s];
        __syncthreads();
    }
    if (threadIdx.x == 0)
        out_loss[0] = (float)NS * red[0] / ((float)NT * (float)NT);
}

extern "C" void kernel_launch(void* const* d_in, const int* in_sizes, int n_in,
                              void* d_out, int out_size, void* d_ws, size_t ws_size,
                              hipStream_t stream) {
    (void)in_sizes; (void)n_in; (void)out_size; (void)ws_size;
    const float* x    = (const float*)d_in[0];   // [B,L,H,D] f32
    const float* w    = (const float*)d_in[1];   // [H,S,D]  f32
    const float* bias = (const float*)d_in[2];   // [H,S]    f32

    float* out      = (float*)d_out;
    float* out_sg   = out;                 // [B,L,H] = 131072 (all 1.0)
    float* out_idx  = out + NT * NH;       // [B,L,H] = 131072 (argmax as value)
    float* out_loss = out + 2 * NT * NH;   // scalar

    unsigned* g_cnt  = (unsigned*)d_ws;                       // [H*S] u32
    float*    g_ssum = (float*)((char*)d_ws + NH * NS * 4);   // [H*S] f32

    router_zero_ws<<<4, 256, 0, stream>>>(g_cnt, g_ssum);
    dim3 grid(NT / ROWS, NH);
    router_main<<<grid, 256, 0, stream>>>(x, w, bias, out_sg, out_idx, g_cnt, g_ssum);
    router_finalize<<<1, 256, 0, stream>>>(g_cnt, g_ssum, out_loss);
}